// OuterProductModule_8899172237442
// MI455X (gfx1250) — compile-verified
//
#include <hip/hip_runtime.h>

typedef __attribute__((ext_vector_type(2))) float v2f;
typedef __attribute__((ext_vector_type(8))) float v8f;

#define BQ 4
#define LL 512
#define PP 512
#define DD 256
#define CC 16
#define OD 32

__device__ __forceinline__ float wave_reduce_sum(float v) {
#pragma unroll
  for (int m = 16; m >= 1; m >>= 1) v += __shfl_xor(v, m, 32);
  return v;
}

// Kernel 1: fused LayerNorm + 16-channel projection.
// One wave32 per row (each lane owns 8 contiguous elements of the 256-dim row).
// Rows 0..2047 -> h_l path, 2048..4095 -> h_p path.
__global__ __launch_bounds__(256) void ln_proj_kernel(
    const float* __restrict__ h_l, const float* __restrict__ h_p,
    const float* __restrict__ ln_l_w, const float* __restrict__ ln_l_b,
    const float* __restrict__ ln_p_w, const float* __restrict__ ln_p_b,
    const float* __restrict__ W_l, const float* __restrict__ W_p,
    float* __restrict__ hl_out, float* __restrict__ hp_out) {
  const int lane = threadIdx.x & 31;
  const int wave = threadIdx.x >> 5;
  const int row  = blockIdx.x * 8 + wave;          // 0..4095
  const bool is_p = row >= (BQ * LL);
  const int  r    = is_p ? row - (BQ * LL) : row;  // b*512 + seq index

  const float* xb = (is_p ? h_p : h_l) + (size_t)r * DD + lane * 8;
  const float* wv = (is_p ? ln_p_w : ln_l_w) + lane * 8;
  const float* bv = (is_p ? ln_p_b : ln_l_b) + lane * 8;
  const float* W  = (is_p ? W_p : W_l);
  float* op = (is_p ? hp_out : hl_out) + (size_t)r * CC;

  float x[8];
  {
    float4 t0 = ((const float4*)xb)[0];
    float4 t1 = ((const float4*)xb)[1];
    x[0]=t0.x; x[1]=t0.y; x[2]=t0.z; x[3]=t0.w;
    x[4]=t1.x; x[5]=t1.y; x[6]=t1.z; x[7]=t1.w;
  }

  float s = 0.f;
#pragma unroll
  for (int k = 0; k < 8; ++k) s += x[k];
  s = wave_reduce_sum(s);
  const float mean = s * (1.0f / DD);

  float v = 0.f;
#pragma unroll
  for (int k = 0; k < 8; ++k) { float d = x[k] - mean; v += d * d; }
  v = wave_reduce_sum(v) * (1.0f / DD);
  const float rstd = rsqrtf(v + 1e-5f);

  float y[8];
  {
    float4 w0 = ((const float4*)wv)[0];
    float4 w1 = ((const float4*)wv)[1];
    float4 b0 = ((const float4*)bv)[0];
    float4 b1 = ((const float4*)bv)[1];
    const float wa[8] = {w0.x,w0.y,w0.z,w0.w,w1.x,w1.y,w1.z,w1.w};
    const float ba[8] = {b0.x,b0.y,b0.z,b0.w,b1.x,b1.y,b1.z,b1.w};
#pragma unroll
    for (int k = 0; k < 8; ++k) y[k] = fmaf((x[k] - mean) * rstd, wa[k], ba[k]);
  }

  // 16 projection channels: dot(y_row, W[c,:]) with wave reduction.
#pragma unroll
  for (int c = 0; c < CC; ++c) {
    const float* Wr = W + (size_t)c * DD + lane * 8;
    float4 a0 = ((const float4*)Wr)[0];
    float4 a1 = ((const float4*)Wr)[1];
    float p = y[0]*a0.x + y[1]*a0.y + y[2]*a0.z + y[3]*a0.w
            + y[4]*a1.x + y[5]*a1.y + y[6]*a1.z + y[7]*a1.w;
    p = wave_reduce_sum(p);
    if (lane == 0) op[c] = p;
  }
}

// Kernel 2: one wave per 16x16 (l,p) tile.
// z-tile = hl_tile @ hp_tile^T via 4x V_WMMA_F32_16X16X4_F32 (K=16 in fp32),
// then expand each z to 32 contiguous outputs: out = z/16 * W_f + b_f.
__global__ __launch_bounds__(128) void outer_expand_kernel(
    const float* __restrict__ hl, const float* __restrict__ hp,
    const float* __restrict__ W_f, const float* __restrict__ b_f,
    float* __restrict__ out) {
  const int lane = threadIdx.x & 31;
  const int wave = threadIdx.x >> 5;
  const int tile = blockIdx.x * 4 + wave;   // 0..4095
  const int pt = tile & 31;                  // p tile
  const int lt = (tile >> 5) & 31;           // l tile
  const int bq = tile >> 10;                 // batch

  const int mn    = lane & 15;   // A: row M  / B: col N
  const int khalf = lane >> 4;   // selects K-pair within each K=4 step

  // A fragment rows come from hl, B fragment (B[k][n] = hp[n][k]) from hp:
  // identical row-major indexing for both.
  const float* arow = hl + ((size_t)bq * LL + lt * 16 + mn) * CC;
  const float* brow = hp + ((size_t)bq * PP + pt * 16 + mn) * CC;

  v8f acc = {};
#pragma unroll
  for (int kk = 0; kk < CC; kk += 4) {
    const int k0 = kk + khalf * 2;
    v2f a; a.x = arow[k0]; a.y = arow[k0 + 1];
    v2f b; b.x = brow[k0]; b.y = brow[k0 + 1];
    acc = __builtin_amdgcn_wmma_f32_16x16x4_f32(
        /*neg_a=*/false, a, /*neg_b=*/false, b,
        /*c_mod=*/(short)0, acc, /*reuse_a=*/false, /*reuse_b=*/false);
  }

  // Broadcast the 32-wide final linear params into registers.
  float4 wf[8], bf[8];
  const float4* Wf4 = (const float4*)W_f;
  const float4* bf4 = (const float4*)b_f;
#pragma unroll
  for (int j = 0; j < 8; ++j) { wf[j] = Wf4[j]; bf[j] = bf4[j]; }

  // C/D layout: VGPR r holds (M = r + 8*khalf, N = mn).
#pragma unroll
  for (int r = 0; r < 8; ++r) {
    const int M = r + khalf * 8;
    const int l = lt * 16 + M;
    const int p = pt * 16 + mn;
    const float z = acc[r] * (1.0f / CC);
    float4* dst = (float4*)(out + ((((size_t)bq * LL + l) * PP) + p) * OD);
#pragma unroll
    for (int j = 0; j < 8; ++j) {
      float4 o;
      o.x = fmaf(z, wf[j].x, bf[j].x);
      o.y = fmaf(z, wf[j].y, bf[j].y);
      o.z = fmaf(z, wf[j].z, bf[j].z);
      o.w = fmaf(z, wf[j].w, bf[j].w);
      dst[j] = o;
    }
  }
}

extern "C" void kernel_launch(void* const* d_in, const int* in_sizes, int n_in,
                              void* d_out, int out_size, void* d_ws, size_t ws_size,
                              hipStream_t stream) {
  const float* h_l    = (const float*)d_in[0];
  const float* h_p    = (const float*)d_in[1];
  const float* ln_l_w = (const float*)d_in[2];
  const float* ln_l_b = (const float*)d_in[3];
  const float* ln_p_w = (const float*)d_in[4];
  const float* ln_p_b = (const float*)d_in[5];
  const float* W_l    = (const float*)d_in[6];
  const float* W_p    = (const float*)d_in[7];
  const float* W_f    = (const float*)d_in[8];
  const float* b_f    = (const float*)d_in[9];

  float* hl = (float*)d_ws;                       // [4,512,16]
  float* hp = hl + (size_t)BQ * LL * CC;          // [4,512,16]

  // 4096 rows total, 8 waves per block of 256 threads.
  ln_proj_kernel<<<(BQ * (LL + PP)) / 8, 256, 0, stream>>>(
      h_l, h_p, ln_l_w, ln_l_b, ln_p_w, ln_p_b, W_l, W_p, hl, hp);

  // 4096 tiles (4 batches x 32 x 32), 4 waves per block of 128 threads.
  outer_expand_kernel<<<(BQ * (LL / 16) * (PP / 16)) / 4, 128, 0, stream>>>(
      hl, hp, W_f, b_f, (float*)d_out);
}